// ROIAlign_4501125726894
// MI455X (gfx1250) — compile-verified
//
#include <hip/hip_runtime.h>
#include <stdint.h>

#define OUT_HW 7
#define C_CH   256
#define F_H    200
#define F_W    200

struct alignas(8) F2 { float x, y; };

// One block = one (roi, ph) row of 7 bins. 256 threads = 256 channels (wave32 x8,
// all lanes active -> async-to-LDS legal). Double-buffered async gather pipeline:
// stage pw+1 is DMA'd into LDS while bin pw is interpolated from LDS.
__global__ __launch_bounds__(256)
void roi_align_async_kernel(const float* __restrict__ feat,
                            const float* __restrict__ rois,
                            float* __restrict__ out)
{
    // [slot][channel(thread)][4 samples * (2 top + 2 bot floats)] = 32 KB
    __shared__ float buf[2][C_CH][16];

    const int tid = threadIdx.x;            // channel
    const int blk = blockIdx.x;             // r*7 + ph
    const int r   = blk / OUT_HW;
    const int ph  = blk - r * OUT_HW;

    // ---- uniform ROI geometry (blockIdx-derived -> scalarizable) ----
    const float* R = rois + r * 5;
    const int   bi  = (int)R[0];
    const float rx1 = R[1] * 0.25f, ry1 = R[2] * 0.25f;
    const float rx2 = R[3] * 0.25f, ry2 = R[4] * 0.25f;
    const float rw  = fmaxf(rx2 - rx1, 1.0f), rh = fmaxf(ry2 - ry1, 1.0f);
    const float bw  = rw * (1.0f / OUT_HW),   bh = rh * (1.0f / OUT_HW);

    // vertical taps for this bin-row (2 sub-samples); clamp folded into (yl, wy)
    int yl[2]; float wy[2];
#pragma unroll
    for (int sy = 0; sy < 2; ++sy) {
        float yy = ry1 + ((float)ph + 0.25f + 0.5f * (float)sy) * bh - 0.5f;
        float py = fminf(fmaxf(yy, 0.0f), (float)(F_H - 1));
        int   y0 = (int)py;                      // py >= 0 -> trunc == floor
        yl[sy]   = (y0 < F_H - 2) ? y0 : (F_H - 2);
        wy[sy]   = py - (float)yl[sy];           // ==1 at clamped bottom edge
    }

    // horizontal taps for all 7 bins
    int xl[OUT_HW][2]; float wx[OUT_HW][2];
#pragma unroll
    for (int pw = 0; pw < OUT_HW; ++pw)
#pragma unroll
        for (int sx = 0; sx < 2; ++sx) {
            float xx = rx1 + ((float)pw + 0.25f + 0.5f * (float)sx) * bw - 0.5f;
            float px = fminf(fmaxf(xx, 0.0f), (float)(F_W - 1));
            int   x0 = (int)px;
            xl[pw][sx] = (x0 < F_W - 2) ? x0 : (F_W - 2);
            wx[pw][sx] = px - (float)xl[pw][sx];
        }

    // byte offset of this (batch, channel) plane; max ~164 MB < 2^31
    const unsigned cbase = (unsigned)((bi * C_CH + tid) * (F_H * F_W)) * 4u;
    // workgroup-relative LDS byte address of this thread's slot 0
    // (generic shared address: low 32 bits == LDS offset)
    const unsigned ldsb  = (unsigned)(size_t)(&buf[0][tid][0]);

    // DMA one bin's 4 samples (top+bottom float2 pairs) into LDS slot.
    auto issue = [&](int pw, int slot) {
        const unsigned lb = ldsb + (unsigned)slot * (unsigned)(C_CH * 16 * 4);
#pragma unroll
        for (int sy = 0; sy < 2; ++sy) {
            const unsigned rowoff = cbase + (unsigned)(yl[sy] * F_W) * 4u;
#pragma unroll
            for (int sx = 0; sx < 2; ++sx) {
                unsigned vo_t = rowoff + (unsigned)xl[pw][sx] * 4u;   // row yl
                unsigned vo_b = vo_t + (unsigned)(F_W * 4);           // row yl+1
                unsigned ld_t = lb + (unsigned)((sy * 2 + sx) * 16);
                unsigned ld_b = ld_t + 8u;
                // GVS form: LDS address in VGPR ld, global address is SGPR feat plus VGPR offset vo
                asm volatile("global_load_async_to_lds_b64 %0, %1, %2"
                             :: "v"(ld_t), "v"(vo_t), "s"(feat) : "memory");
                asm volatile("global_load_async_to_lds_b64 %0, %1, %2"
                             :: "v"(ld_b), "v"(vo_b), "s"(feat) : "memory");
            }
        }
    };

    auto compute_bin = [&](int pw, int slot) -> float {
        const float* s = &buf[slot][tid][0];
        float acc = 0.0f;
#pragma unroll
        for (int k = 0; k < 4; ++k) {            // k = sy*2 + sx
            F2 t = *(const F2*)(s + k * 4);
            F2 b = *(const F2*)(s + k * 4 + 2);
            float wxx = wx[pw][k & 1];
            float wyy = wy[k >> 1];
            float top = fmaf(wxx, t.y - t.x, t.x);
            float bot = fmaf(wxx, b.y - b.x, b.x);
            acc += fmaf(wyy, bot - top, top);
        }
        return acc * 0.25f;
    };

    // ---- 2-deep software pipeline over the 7 bins ----
    issue(0, 0);
    issue(1, 1);

    const size_t obase = ((size_t)r * C_CH + tid) * (OUT_HW * OUT_HW)
                       + (size_t)ph * OUT_HW;

#pragma unroll
    for (int pw = 0; pw < OUT_HW; ++pw) {
        if (pw < OUT_HW - 1)
            asm volatile("s_wait_asynccnt 8" ::: "memory");  // stage pw retired
        else
            asm volatile("s_wait_asynccnt 0" ::: "memory");  // last stage
        float v = compute_bin(pw, pw & 1);
        out[obase + pw] = v;
        if (pw + 2 < OUT_HW) {
            // close the ds-read vs async-LDS-write WAR window before slot reuse
            asm volatile("s_wait_dscnt 0" ::: "memory");
            issue(pw + 2, pw & 1);
        }
    }
}

extern "C" void kernel_launch(void* const* d_in, const int* in_sizes, int n_in,
                              void* d_out, int out_size, void* d_ws, size_t ws_size,
                              hipStream_t stream) {
    (void)n_in; (void)out_size; (void)d_ws; (void)ws_size;
    const float* feat = (const float*)d_in[0];
    const float* rois = (const float*)d_in[1];
    float* out = (float*)d_out;
    const int B = in_sizes[1] / 5;          // number of ROIs
    dim3 grid(B * OUT_HW);
    dim3 block(C_CH);
    roi_align_async_kernel<<<grid, block, 0, stream>>>(feat, rois, out);
}